// GAT_27092653703958
// MI455X (gfx1250) — compile-verified
//
#include <hip/hip_runtime.h>
#include <math.h>

#define F_IN 128
#define HID  64
#define NEG_SLOPE 0.2f

typedef __attribute__((ext_vector_type(2))) float v2f;
typedef __attribute__((ext_vector_type(8))) float v8f;

// ---------- helpers ----------
__device__ __forceinline__ float lrelu(float x) {
    return x >= 0.f ? x : NEG_SLOPE * x;
}

// float atomic max via int/uint ordering trick (lowers to native int atomics)
__device__ __forceinline__ void atomicMaxF(float* addr, float val) {
    if (val >= 0.f) {
        atomicMax((int*)addr, __float_as_int(val));
    } else {
        atomicMin((unsigned int*)addr, __float_as_uint(val));
    }
}

__device__ __forceinline__ void atomicAddF(float* addr, float val) {
    unsafeAtomicAdd(addr, val);   // native global_atomic_add_f32
}

// ---------- init workspace each call (graph-replay safe) ----------
__global__ void gat_init(float* acc1, float* out, float* emax1, float* den1,
                         float* emax2, float* den2, int N) {
    int i = blockIdx.x * blockDim.x + threadIdx.x;
    long tot = (long)N * HID;
    if (i < tot) acc1[i] = 0.f;
    if (i < N * 2) out[i] = 0.f;
    if (i < N) {
        emax1[i] = -INFINITY; den1[i] = 0.f;
        emax2[i] = -INFINITY; den2[i] = 0.f;
    }
}

// ---------- GEMM1: h1[N,64] = x[N,128] @ W1[128,64] via V_WMMA_F32_16X16X4_F32
// block = 128 threads = 4 waves; block handles 16 nodes; wave w handles cols 16w..16w+15
#define XS_LD  132   // 128 + 4 pad (stride mod 64 banks = 4 -> conflict free)
#define WL_LD  68    // 64 + 4 pad
__global__ void gat_gemm1_wmma(const float* __restrict__ x,
                               const float* __restrict__ W1,
                               float* __restrict__ h1, int N) {
    __shared__ float xs[16 * XS_LD];
    __shared__ float wl[F_IN * WL_LD];

    const int tid  = threadIdx.x;
    const int m0   = blockIdx.x * 16;

    // stage x tile (16 x 128)
    for (int i = tid; i < 16 * F_IN; i += 128) {
        int r = i >> 7, k = i & 127;
        int row = m0 + r;
        xs[r * XS_LD + k] = (row < N) ? x[(size_t)row * F_IN + k] : 0.f;
    }
    // stage W1 (128 x 64), coalesced
    for (int i = tid; i < F_IN * HID; i += 128) {
        int k = i >> 6, n = i & 63;
        wl[k * WL_LD + n] = W1[i];
    }
    __syncthreads();

    const int wave = tid >> 5;
    const int lane = tid & 31;
    const int half = lane >> 4;      // 0: lanes 0-15, 1: lanes 16-31
    const int lm   = lane & 15;
    const int n0   = wave * 16;

    v8f c = {};
    #pragma unroll
    for (int kk = 0; kk < F_IN; kk += 4) {
        const int kb = kk + half * 2;
        // A frag: lane holds M=lm, K=kb..kb+1 (consecutive -> float2 LDS load)
        v2f a;
        const float2 av = *(const float2*)&xs[lm * XS_LD + kb];
        a.x = av.x; a.y = av.y;
        // B frag: lane holds N=lm, K=kb..kb+1
        v2f b;
        b.x = wl[(kb + 0) * WL_LD + n0 + lm];
        b.y = wl[(kb + 1) * WL_LD + n0 + lm];
        c = __builtin_amdgcn_wmma_f32_16x16x4_f32(false, a, false, b,
                                                  (short)0, c, false, false);
    }

    // C layout: VGPR r -> row = r + half*8, col = n0 + lm
    #pragma unroll
    for (int r = 0; r < 8; ++r) {
        int row = m0 + r + half * 8;
        if (row < N) h1[(size_t)row * HID + n0 + lm] = c[r];
    }
}

// ---------- per-node attention logits: as[n]=h.a_src, ad[n]=h.a_dst (wave/node)
__global__ void gat_alpha64(const float* __restrict__ h,
                            const float* __restrict__ a_src,
                            const float* __restrict__ a_dst,
                            float* __restrict__ as, float* __restrict__ ad, int N) {
    int node = blockIdx.x * 8 + (threadIdx.x >> 5);
    int lane = threadIdx.x & 31;
    if (node >= N) return;
    const float* row = h + (size_t)node * HID;
    float v0 = row[lane], v1 = row[lane + 32];
    float ps = v0 * a_src[lane] + v1 * a_src[lane + 32];
    float pd = v0 * a_dst[lane] + v1 * a_dst[lane + 32];
    #pragma unroll
    for (int off = 16; off; off >>= 1) {
        ps += __shfl_down(ps, off);
        pd += __shfl_down(pd, off);
    }
    if (lane == 0) { as[node] = ps; ad[node] = pd; }
}

// ---------- message src/dst (edges + self loops) ----------
__device__ __forceinline__ void msg_sd(const int* ei, int E, int m, int& s, int& d) {
    if (m < E) { s = ei[m]; d = ei[E + m]; }
    else       { s = d = m - E; }
}

// pass 1: segment max of leaky_relu(as[src]+ad[dst]) over dst
__global__ void gat_edge_max(const int* __restrict__ ei, int E, int N,
                             const float* __restrict__ as, const float* __restrict__ ad,
                             float* __restrict__ emax) {
    int m = blockIdx.x * blockDim.x + threadIdx.x;
    if (m >= E + N) return;
    int s, d; msg_sd(ei, E, m, s, d);
    atomicMaxF(&emax[d], lrelu(as[s] + ad[d]));
}

// pass 2: denom[dst] += exp(e - emax[dst])
__global__ void gat_edge_sum(const int* __restrict__ ei, int E, int N,
                             const float* __restrict__ as, const float* __restrict__ ad,
                             const float* __restrict__ emax, float* __restrict__ den) {
    int m = blockIdx.x * blockDim.x + threadIdx.x;
    if (m >= E + N) return;
    int s, d; msg_sd(ei, E, m, s, d);
    float e = lrelu(as[s] + ad[d]);
    atomicAddF(&den[d], expf(e - emax[d]));
}

// pass 3 (layer 1): acc[dst,:] += alpha * h[src,:]  (wave per message, 64 ch)
__global__ void gat_edge_scatter64(const int* __restrict__ ei, int E, int N,
                                   const float* __restrict__ as, const float* __restrict__ ad,
                                   const float* __restrict__ emax, const float* __restrict__ den,
                                   const float* __restrict__ h, float* __restrict__ acc) {
    int wid  = (blockIdx.x * blockDim.x + threadIdx.x) >> 5;
    int lane = threadIdx.x & 31;
    if (wid >= E + N) return;
    int s, d; msg_sd(ei, E, wid, s, d);
    float w = 0.f;
    if (lane == 0) {
        float e = lrelu(as[s] + ad[d]);
        w = expf(e - emax[d]) / den[d];
    }
    w = __shfl(w, 0);
    const float* hs = h + (size_t)s * HID;
    float* ac = acc + (size_t)d * HID;
    atomicAddF(&ac[lane],      w * hs[lane]);
    atomicAddF(&ac[lane + 32], w * hs[lane + 32]);
}

// ---------- fused: v = relu(acc1+b1); h2 = v @ W2[64,2]; as2/ad2 dots (wave/node)
__global__ void gat_layer2_feat(const float* __restrict__ acc1, const float* __restrict__ b1,
                                const float* __restrict__ W2,
                                const float* __restrict__ a2s, const float* __restrict__ a2d,
                                float* __restrict__ h2, float* __restrict__ as2,
                                float* __restrict__ ad2, int N) {
    int node = blockIdx.x * 8 + (threadIdx.x >> 5);
    int lane = threadIdx.x & 31;
    if (node >= N) return;
    const float* row = acc1 + (size_t)node * HID;
    float v0 = fmaxf(row[lane]      + b1[lane],      0.f);
    float v1 = fmaxf(row[lane + 32] + b1[lane + 32], 0.f);
    float p0 = v0 * W2[lane * 2 + 0] + v1 * W2[(lane + 32) * 2 + 0];
    float p1 = v0 * W2[lane * 2 + 1] + v1 * W2[(lane + 32) * 2 + 1];
    #pragma unroll
    for (int off = 16; off; off >>= 1) {
        p0 += __shfl_down(p0, off);
        p1 += __shfl_down(p1, off);
    }
    if (lane == 0) {
        h2[node * 2 + 0] = p0;
        h2[node * 2 + 1] = p1;
        as2[node] = p0 * a2s[0] + p1 * a2s[1];
        ad2[node] = p0 * a2d[0] + p1 * a2d[1];
    }
}

// pass 3 (layer 2): out[dst,0:2] += alpha * h2[src,0:2]  (thread per message)
__global__ void gat_edge_scatter2(const int* __restrict__ ei, int E, int N,
                                  const float* __restrict__ as, const float* __restrict__ ad,
                                  const float* __restrict__ emax, const float* __restrict__ den,
                                  const float* __restrict__ h2, float* __restrict__ out) {
    int m = blockIdx.x * blockDim.x + threadIdx.x;
    if (m >= E + N) return;
    int s, d; msg_sd(ei, E, m, s, d);
    float e = lrelu(as[s] + ad[d]);
    float w = expf(e - emax[d]) / den[d];
    atomicAddF(&out[d * 2 + 0], w * h2[s * 2 + 0]);
    atomicAddF(&out[d * 2 + 1], w * h2[s * 2 + 1]);
}

// final bias add
__global__ void gat_bias2(float* __restrict__ out, const float* __restrict__ b2, int N) {
    int i = blockIdx.x * blockDim.x + threadIdx.x;
    if (i < 2 * N) out[i] += b2[i & 1];
}

extern "C" void kernel_launch(void* const* d_in, const int* in_sizes, int n_in,
                              void* d_out, int out_size, void* d_ws, size_t ws_size,
                              hipStream_t stream) {
    const float* x    = (const float*)d_in[0];
    const int*   ei   = (const int*)  d_in[1];
    const float* W1   = (const float*)d_in[2];
    const float* a1s  = (const float*)d_in[3];
    const float* a1d  = (const float*)d_in[4];
    const float* b1   = (const float*)d_in[5];
    const float* W2   = (const float*)d_in[6];
    const float* a2s  = (const float*)d_in[7];
    const float* a2d  = (const float*)d_in[8];
    const float* b2   = (const float*)d_in[9];
    float* out = (float*)d_out;

    const int N = in_sizes[0] / F_IN;
    const int E = in_sizes[1] / 2;
    const int M = E + N;   // messages incl. self loops

    // workspace carve-up (all fp32)
    float* h1    = (float*)d_ws;                 // N*64
    float* acc1  = h1    + (size_t)N * HID;      // N*64
    float* as1   = acc1  + (size_t)N * HID;      // N
    float* ad1   = as1   + N;
    float* emax1 = ad1   + N;
    float* den1  = emax1 + N;
    float* h2    = den1  + N;                    // N*2
    float* as2   = h2    + (size_t)N * 2;
    float* ad2   = as2   + N;
    float* emax2 = ad2   + N;
    float* den2  = emax2 + N;

    const int T = 256;
    // init
    {
        long tot = (long)N * HID;
        gat_init<<<(int)((tot + T - 1) / T), T, 0, stream>>>(acc1, out, emax1, den1,
                                                             emax2, den2, N);
    }
    // layer 1 GEMM (WMMA) + attention logits
    gat_gemm1_wmma<<<(N + 15) / 16, 128, 0, stream>>>(x, W1, h1, N);
    gat_alpha64<<<(N + 7) / 8, T, 0, stream>>>(h1, a1s, a1d, as1, ad1, N);
    // layer 1 segment softmax + scatter
    gat_edge_max<<<(M + T - 1) / T, T, 0, stream>>>(ei, E, N, as1, ad1, emax1);
    gat_edge_sum<<<(M + T - 1) / T, T, 0, stream>>>(ei, E, N, as1, ad1, emax1, den1);
    gat_edge_scatter64<<<(M + 7) / 8, T, 0, stream>>>(ei, E, N, as1, ad1, emax1, den1,
                                                      h1, acc1);
    // layer 2 features + logits
    gat_layer2_feat<<<(N + 7) / 8, T, 0, stream>>>(acc1, b1, W2, a2s, a2d,
                                                   h2, as2, ad2, N);
    // layer 2 segment softmax + scatter
    gat_edge_max<<<(M + T - 1) / T, T, 0, stream>>>(ei, E, N, as2, ad2, emax2);
    gat_edge_sum<<<(M + T - 1) / T, T, 0, stream>>>(ei, E, N, as2, ad2, emax2, den2);
    gat_edge_scatter2<<<(M + T - 1) / T, T, 0, stream>>>(ei, E, N, as2, ad2, emax2, den2,
                                                         h2, out);
    gat_bias2<<<(2 * N + T - 1) / T, T, 0, stream>>>(out, b2, N);
}